// SoftDTW_22127671509409
// MI455X (gfx1250) — compile-verified
//
#include <hip/hip_runtime.h>
#include <stdint.h>

// Problem constants (fixed by the reference: N = M = 4096, GAMMA = 1).
#define SN 4096
#define STRIP_ROWS 32                 // one wave32 per strip: lane t owns row 32*s + t
#define NSTRIPS (SN / STRIP_ROWS)     // 128 strips
#define CH 32                         // systolic chunk (columns per sync window)
#define INF_F 3.0e38f

// ---------------------------------------------------------------------------
// CDNA5 async global->LDS copy (GLOBAL_LOAD_ASYNC_TO_LDS_B32, ASYNCcnt path).
// Builtin signature (from hipcc diagnostic): arg0 = global int*, arg1 = LDS
// int*, then (offset, cpol) immediates.
// ---------------------------------------------------------------------------
typedef __attribute__((address_space(1))) int* gptr_i32;
typedef __attribute__((address_space(3))) int* lptr_i32;

__device__ __forceinline__ void async_g2l_b32(const float* gp, float* lp) {
#if __has_builtin(__builtin_amdgcn_global_load_async_to_lds_b32)
  __builtin_amdgcn_global_load_async_to_lds_b32(
      (gptr_i32)(uintptr_t)gp,
      (lptr_i32)lp,
      /*offset=*/0, /*cpol=*/0);
#else
  uint32_t loff = (uint32_t)(uintptr_t)lp;   // low 32 bits of generic LDS ptr = LDS byte offset
  asm volatile("global_load_async_to_lds_b32 %0, %1, off"
               :: "v"(loff), "v"(gp)
               : "memory");
#endif
}

__device__ __forceinline__ void wait_asynccnt0() {
#if __has_builtin(__builtin_amdgcn_s_wait_asynccnt)
  __builtin_amdgcn_s_wait_asynccnt(0);
#else
  asm volatile("s_wait_asynccnt 0" ::: "memory");
#endif
}

// ---------------------------------------------------------------------------
// Probe kernel (FIRST in file, never launched): minimal body so the compile
// loop's asm snippet directly shows the async global->LDS instruction and
// s_wait_asynccnt that the helpers lower to.
// ---------------------------------------------------------------------------
__global__ void cdna5_async_probe(const float* __restrict__ src,
                                  float* __restrict__ dst) {
  __shared__ float buf[32];
  async_g2l_b32(src + threadIdx.x, &buf[threadIdx.x]);
  wait_asynccnt0();
  __syncthreads();
  dst[threadIdx.x] = buf[threadIdx.x];
}

// ---------------------------------------------------------------------------
// Kernel 2: acc_grad stencil, bandwidth-bound (~128 MB @ 23.3 TB/s -> ~5.5 us).
//   d = exp(-exp(-D));  grad[i,j] = d[i+1,j+1] + d[i+1,j] + d[i,j+1] - d[i,j]
// Tile (+1 halo) of raw D is pulled into LDS with async global->LDS copies,
// converted to d in place (1 exp-pair per element), then the 2x2 stencil is
// read from LDS. Interior blocks (all but last block row/col) skip all bounds
// predication.
// ---------------------------------------------------------------------------
#define TX 32
#define TY 8
__global__ __launch_bounds__(256) void softdtw_grad(const float* __restrict__ D,
                                                    float* __restrict__ grad) {
  __shared__ float tile[(TY + 1) * (TX + 4)];  // 9 x 36 (padded stride)

  const int j0  = blockIdx.x * TX;
  const int i0  = blockIdx.y * TY;
  const int tid = threadIdx.x;                 // 0..255
  const int NE  = (TY + 1) * (TX + 1);         // 297 entries incl. halo
  const bool interior = (i0 + TY < SN) && (j0 + TX < SN);

  if (interior) {
    // Stage 1: async-copy raw D tile (+halo) into LDS, no predication.
    for (int e = tid; e < NE; e += 256) {
      const int r = e / (TX + 1);
      const int c = e - r * (TX + 1);
      async_g2l_b32(D + (size_t)(i0 + r) * SN + (j0 + c),
                    &tile[r * (TX + 4) + c]);
    }
    wait_asynccnt0();   // each wave drains its own ASYNCcnt ...
    __syncthreads();    // ... then the barrier publishes all LDS writes

    // Stage 2: in-place d = exp(-exp(-D)).
    for (int e = tid; e < NE; e += 256) {
      const int r  = e / (TX + 1);
      const int c  = e - r * (TX + 1);
      const int lo = r * (TX + 4) + c;
      tile[lo] = __expf(-__expf(-tile[lo]));
    }
  } else {
    // Boundary blocks: predicated loads; out-of-range halo -> d = 0 (zero pad).
    for (int e = tid; e < NE; e += 256) {
      const int r = e / (TX + 1);
      const int c = e - r * (TX + 1);
      const int gi = i0 + r, gj = j0 + c;
      if (gi < SN && gj < SN) {
        async_g2l_b32(D + (size_t)gi * SN + gj, &tile[r * (TX + 4) + c]);
      }
    }
    wait_asynccnt0();
    __syncthreads();
    for (int e = tid; e < NE; e += 256) {
      const int r  = e / (TX + 1);
      const int c  = e - r * (TX + 1);
      const int gi = i0 + r, gj = j0 + c;
      const int lo = r * (TX + 4) + c;
      float v = 0.0f;
      if (gi < SN && gj < SN) v = __expf(-__expf(-tile[lo]));
      tile[lo] = v;
    }
  }
  __syncthreads();

  // Stage 3: 2x2 stencil from LDS, coalesced store.
  const int tx = tid & (TX - 1);
  const int ty = tid >> 5;
  const int lo = ty * (TX + 4) + tx;
  const float d00 = tile[lo];
  const float d01 = tile[lo + 1];
  const float d10 = tile[lo + (TX + 4)];
  const float d11 = tile[lo + (TX + 4) + 1];
  grad[(size_t)(i0 + ty) * SN + (j0 + tx)] = d11 + d10 + d01 - d00;
}

// ---------------------------------------------------------------------------
// Kernel 0: reset inter-strip progress flags (graph-replay safe).
// ---------------------------------------------------------------------------
__global__ void softdtw_init(int* __restrict__ progress) {
  int t = threadIdx.x;
  if (t < NSTRIPS) progress[t] = 0;
}

// ---------------------------------------------------------------------------
// Kernel 1: hard-min DTW DP, systolic wave32 per 32-row strip.
//   c[i,j] = exp(-D[i,j]) + min(c[i-1,j], c[i,j-1], c[i-1,j-1])
// Lane t lags lane t-1 by one column; c flows down via __shfl_up each step.
// Strip s streams its bottom row into rowbuf[s] and publishes progress with
// release semantics; strip s+1 spins with acquire loads (agent scope).
// Pipeline depth ~2 chunks/strip: total ~4127 + 127*64 ~ 12.3k systolic steps.
// ---------------------------------------------------------------------------
__global__ __launch_bounds__(32) void softdtw_dp(const float* __restrict__ D,
                                                 float* __restrict__ cost_out,
                                                 float* __restrict__ rowbuf,
                                                 int* __restrict__ progress) {
  const int s    = blockIdx.x;
  const int lane = threadIdx.x;            // 0..31, wave32
  const int row  = s * STRIP_ROWS + lane;  // always < SN (128*32 == 4096)

  const float* __restrict__ drow  = D + (size_t)row * SN;
  float*       __restrict__ myrow = rowbuf + (size_t)s * SN;
  const float* __restrict__ uprow = rowbuf + (size_t)(s - 1) * SN;

  // Virtual borders: c[-1,-1] = 0 (only entry for cell (0,0)), all other
  // out-of-range c = +INF -> reproduces cumsum first row / first column.
  float left  = INF_F;
  float diag  = (s == 0 && lane == 0) ? 0.0f : INF_F;
  float c_cur = INF_F;

  const int nsteps  = SN + STRIP_ROWS - 1;       // 4127 skewed steps
  const int nchunks = (nsteps + CH - 1) / CH;    // 129 chunks

  for (int ck = 0; ck < nchunks; ++ck) {
    const int k0 = ck * CH;

    // Wait until the strip above has published top-row columns [k0, k0+CH).
    if (s > 0) {
      const int need = (k0 + CH < SN) ? (k0 + CH) : SN;
      while (__hip_atomic_load(&progress[s - 1], __ATOMIC_ACQUIRE,
                               __HIP_MEMORY_SCOPE_AGENT) < need) {
        __builtin_amdgcn_s_sleep(2);
      }
    }

    // Cooperative load of this chunk's top-row values (broadcast per step).
    const int  tcol = k0 + lane;
    const float tv  = (s > 0 && tcol < SN) ? uprow[tcol] : INF_F;

    // Fetch this lane's D values for the chunk and exponentiate off the
    // critical path (lane t covers columns k0-t .. k0-t+CH-1).
    float ev[CH];
#pragma unroll
    for (int st = 0; st < CH; ++st) {
      const int j    = k0 + st - lane;
      const float dv = (j >= 0 && j < SN) ? drow[j] : 0.0f;
      ev[st] = __expf(-dv);
    }

    // Prefetch next chunk's D for this lane (global_prefetch_b8).
    {
      int jn = k0 + CH - lane;
      if (jn < 0) jn = 0;
      if (jn > SN - 1) jn = SN - 1;
      __builtin_prefetch(drow + jn, 0, 0);
      int jn2 = jn + 16;
      if (jn2 > SN - 1) jn2 = SN - 1;
      __builtin_prefetch(drow + jn2, 0, 0);
    }

    // Systolic steps: one column per lane per step.
#pragma unroll
    for (int st = 0; st < CH; ++st) {
      const int k = k0 + st;
      float upin     = __shfl_up(c_cur, 1, STRIP_ROWS);  // c[row-1, j] from lane-1
      const float bc = __shfl(tv, st, STRIP_ROWS);       // top row for lane 0
      if (lane == 0) upin = bc;
      const int j = k - lane;
      if (j >= 0 && j < SN) {
        const float m = fminf(upin, fminf(left, diag));
        const float c = ev[st] + m;
        diag  = upin;   // becomes c[row-1, j] for column j+1
        left  = c;
        c_cur = c;
        if (lane == STRIP_ROWS - 1) myrow[j] = c;        // stream bottom row
      }
    }

    // Publish: after chunk ck, bottom row columns [0, k0] are written.
    if (lane == STRIP_ROWS - 1) {
      const int done = (k0 + 1 < SN) ? (k0 + 1) : SN;
      __hip_atomic_store(&progress[s], done, __ATOMIC_RELEASE,
                         __HIP_MEMORY_SCOPE_AGENT);
    }
  }

  if (s == NSTRIPS - 1 && lane == STRIP_ROWS - 1) {
    cost_out[0] = c_cur;   // c[N-1, M-1]
  }
}

// ---------------------------------------------------------------------------
// Host launcher. d_out = [cost (1 float)] ++ [acc_grad (SN*SN floats)].
// Scratch (DP row buffer + progress flags) lives inside the acc_grad region,
// which the stencil kernel fully overwrites afterwards (stream-ordered), so
// every call is deterministic and graph-capture safe.
// ---------------------------------------------------------------------------
extern "C" void kernel_launch(void* const* d_in, const int* in_sizes, int n_in,
                              void* d_out, int out_size, void* d_ws, size_t ws_size,
                              hipStream_t stream) {
  (void)in_sizes; (void)n_in; (void)d_ws; (void)ws_size; (void)out_size;

  const float* D   = (const float*)d_in[0];
  float* out       = (float*)d_out;
  float* grad      = out + 1;
  float* rowbuf    = grad;                               // first 2 MB of grad region
  int*   progress  = (int*)(grad + (size_t)SN * SN - NSTRIPS);  // tail of grad region

  hipLaunchKernelGGL(softdtw_init, dim3(1), dim3(NSTRIPS), 0, stream, progress);
  hipLaunchKernelGGL(softdtw_dp, dim3(NSTRIPS), dim3(STRIP_ROWS), 0, stream,
                     D, out, rowbuf, progress);
  hipLaunchKernelGGL(softdtw_grad, dim3(SN / TX, SN / TY), dim3(256), 0, stream,
                     D, grad);
}